// ClassLoss_72421738545222
// MI455X (gfx1250) — compile-verified
//
#include <hip/hip_runtime.h>
#include <math.h>

// Problem constants (from the reference): B=32, A=3, H=64, W=64, NC=80, T=50
#define NB 32
#define NA 3
#define NH 64
#define NW 64
#define NC 80
#define NT 50
#define NTGT (NB * NT)        // 1600 targets total
#define STRIDE_C 85           // 5 + NC channels in output tensor

typedef __attribute__((ext_vector_type(2))) float v2f;
typedef __attribute__((ext_vector_type(8))) float v8f;

// ---------------------------------------------------------------------------
// Kernel 1: one wave32 per target.
//   - choose best anchor by IoU (first-max wins, matching jnp.argmax)
//   - gather the 80-class logit row, wave-reduce log-softmax CE
//   - lane 0 writes masked CE and mask to workspace
// Latency-bound gather: 1600 independent waves give maximal MLP on HBM/L2.
// ---------------------------------------------------------------------------
__global__ __launch_bounds__(256) void ce_kernel(
    const float* __restrict__ out,      // (B, A, H, W, 85)
    const float* __restrict__ anchors,  // (A, 2)
    const float* __restrict__ targets,  // (B, T, 5)
    float* __restrict__ ws_ce,          // [NTGT]
    float* __restrict__ ws_mask)        // [NTGT]
{
    const int gid  = blockIdx.x * blockDim.x + threadIdx.x;
    const int wave = gid >> 5;   // target index
    const int lane = gid & 31;
    if (wave >= NTGT) return;

    const int b = wave / NT;

    // Load the 5 target scalars (broadcast through cache; all lanes same addr).
    const float* tg = targets + (size_t)wave * 5;
    const float tc0 = tg[0], tc1 = tg[1], tc2 = tg[2], tc3 = tg[3], tc4 = tg[4];

    int cls = (int)tc0;
    cls = cls < 0 ? 0 : (cls > NC - 1 ? NC - 1 : cls);
    int ti = (int)(tc1 * (float)NW);
    ti = ti < 0 ? 0 : (ti > NW - 1 ? NW - 1 : ti);
    int tj = (int)(tc2 * (float)NH);
    tj = tj < 0 ? 0 : (tj > NH - 1 ? NH - 1 : tj);
    const float tw = tc3 * (float)NW;
    const float th = tc4 * (float)NH;

    // Best anchor by IoU (first occurrence of max, like jnp.argmax).
    float best = -1.0f;
    int best_a = 0;
#pragma unroll
    for (int a = 0; a < NA; ++a) {
        const float aw = anchors[2 * a + 0];
        const float ah = anchors[2 * a + 1];
        const float inter = fminf(aw, tw) * fminf(ah, th);
        const float uni   = aw * ah + tw * th - inter;
        const float iou   = inter / uni;
        if (iou > best) { best = iou; best_a = a; }
    }
    const float mask = (best > 0.5f) ? 1.0f : 0.0f;

    // Gather the 80 logits at (b, best_a, tj, ti, 5:85).
    const size_t row = (size_t)(((b * NA + best_a) * NH + tj) * NW + ti);
    const float* base = out + row * STRIDE_C + 5;

    const float x0 = base[lane];
    const float x1 = base[lane + 32];
    const float x2 = (lane < 16) ? base[lane + 64] : -INFINITY;

    // Wave32 max reduction.
    float m = fmaxf(x0, fmaxf(x1, x2));
#pragma unroll
    for (int off = 16; off > 0; off >>= 1)
        m = fmaxf(m, __shfl_xor(m, off, 32));

    // Wave32 sum of exp(x - m).
    float s = expf(x0 - m) + expf(x1 - m) + ((lane < 16) ? expf(x2 - m) : 0.0f);
#pragma unroll
    for (int off = 16; off > 0; off >>= 1)
        s += __shfl_xor(s, off, 32);

    if (lane == 0) {
        const float x_cls = base[cls];
        const float ce = logf(s) + m - x_cls;   // -(x_cls - m - log(sum))
        ws_ce[wave]   = ce * mask;
        ws_mask[wave] = mask;
    }
}

// ---------------------------------------------------------------------------
// Kernel 2: single wave32, deterministic final reduction via WMMA.
// D = A(16x4) * ones(4x16) + C  => each row of D holds rowsum(A), replicated
// across the 16 columns. Accumulating 25 chunks of 64 values in C, then
// reducing all lanes/components and dividing by 16 (exact, power of two)
// yields the exact f32 sum in a fixed order => bit-deterministic replays.
// EXEC is all-1s (exactly 32 threads launched), as WMMA requires.
// A-matrix 16x4 f32 layout: lanes 0-15 hold (M=lane, K=0..1) in v[0..1],
// lanes 16-31 hold (M=lane-16, K=2..3). B is all-ones so its layout is moot.
// ---------------------------------------------------------------------------
__global__ __launch_bounds__(32) void reduce_wmma_kernel(
    const float* __restrict__ ws_ce,
    const float* __restrict__ ws_mask,
    float* __restrict__ out)
{
    const int lane = threadIdx.x;
    const int M  = lane & 15;
    const int kb = (lane >> 4) << 1;   // 0 for lanes 0-15, 2 for lanes 16-31

    v8f acc_ce = {0.f, 0.f, 0.f, 0.f, 0.f, 0.f, 0.f, 0.f};
    v8f acc_m  = {0.f, 0.f, 0.f, 0.f, 0.f, 0.f, 0.f, 0.f};
    v2f ones;
    ones.x = 1.0f; ones.y = 1.0f;

#pragma unroll 1
    for (int chunk = 0; chunk < NTGT / 64; ++chunk) {   // 25 chunks of 64
        const int i0 = chunk * 64 + M * 4 + kb;
        v2f a, bm;
        a.x  = ws_ce[i0];     a.y  = ws_ce[i0 + 1];
        bm.x = ws_mask[i0];   bm.y = ws_mask[i0 + 1];
        acc_ce = __builtin_amdgcn_wmma_f32_16x16x4_f32(
            /*neg_a=*/false, a,  /*neg_b=*/false, ones,
            /*c_mod=*/(short)0, acc_ce, /*reuse_a=*/false, /*reuse_b=*/false);
        acc_m  = __builtin_amdgcn_wmma_f32_16x16x4_f32(
            /*neg_a=*/false, bm, /*neg_b=*/false, ones,
            /*c_mod=*/(short)0, acc_m,  /*reuse_a=*/false, /*reuse_b=*/false);
    }

    // Sum this lane's 8 D-components, then full-wave reduce; each column of D
    // holds the complete rowsum vector, so grand total = reduce / 16 (exact).
    float sc = acc_ce[0] + acc_ce[1] + acc_ce[2] + acc_ce[3] +
               acc_ce[4] + acc_ce[5] + acc_ce[6] + acc_ce[7];
    float sm = acc_m[0] + acc_m[1] + acc_m[2] + acc_m[3] +
               acc_m[4] + acc_m[5] + acc_m[6] + acc_m[7];
#pragma unroll
    for (int off = 16; off > 0; off >>= 1) {
        sc += __shfl_xor(sc, off, 32);
        sm += __shfl_xor(sm, off, 32);
    }
    sc *= (1.0f / 16.0f);
    sm *= (1.0f / 16.0f);

    if (lane == 0)
        out[0] = (sm > 0.0f) ? (sc / sm) : 0.0f;
}

extern "C" void kernel_launch(void* const* d_in, const int* in_sizes, int n_in,
                              void* d_out, int out_size, void* d_ws, size_t ws_size,
                              hipStream_t stream) {
    (void)in_sizes; (void)n_in; (void)out_size; (void)ws_size;
    const float* output  = (const float*)d_in[0];   // (32,3,64,64,85) f32
    const float* anchors = (const float*)d_in[1];   // (3,2) f32
    const float* targets = (const float*)d_in[2];   // (32,50,5) f32

    float* ws_ce   = (float*)d_ws;        // 1600 floats
    float* ws_mask = ws_ce + NTGT;        // 1600 floats  (12.8 KB total)

    // 1600 waves, one per target: 200 blocks x 256 threads (8 waves/block).
    ce_kernel<<<(NTGT * 32 + 255) / 256, 256, 0, stream>>>(
        output, anchors, targets, ws_ce, ws_mask);

    // One full wave for the exact WMMA-based reduction + epilogue.
    reduce_wmma_kernel<<<1, 32, 0, stream>>>(ws_ce, ws_mask, (float*)d_out);
}